// Router_90795608637904
// MI455X (gfx1250) — compile-verified
//
#include <hip/hip_runtime.h>

typedef float v2f __attribute__((ext_vector_type(2)));
typedef float v4f __attribute__((ext_vector_type(4)));
typedef float v8f __attribute__((ext_vector_type(8)));

#define NTOK   8192      // B*T
#define NEMBD  1024
#define NEXP   8
#define CAP    512
#define OUT_HALF 33554432  // B*T*E*C

// ---------------------------------------------------------------------------
// K1: gate logits GEMM via V_WMMA_F32_16X16X4_F32.
// One wave computes a 16-token x 16-expert tile (experts 8..15 are zero pad).
// ---------------------------------------------------------------------------
__global__ void __launch_bounds__(128)
gate_gemm_kernel(const float* __restrict__ x, const float* __restrict__ wg,
                 float* __restrict__ logits) {
    const int wave = (blockIdx.x * blockDim.x + threadIdx.x) >> 5;   // 0..511
    const int lane = threadIdx.x & 31;
    const int l    = lane & 15;           // M row (A) / N col (B,C)
    const int h    = lane >> 4;           // K-pair selector
    const int rowBase = wave * 16;

    const float* arow = x + (size_t)(rowBase + l) * NEMBD + 2 * h;
    // branchless zero-pad for experts 8..15: clamp row, mask value
    const int   bl   = (l < NEXP) ? l : (NEXP - 1);
    const float bm   = (l < NEXP) ? 1.0f : 0.0f;
    const float* brow = wg + (size_t)bl * NEMBD + 2 * h;

    v8f c = {};
#pragma unroll 4
    for (int k0 = 0; k0 < NEMBD; k0 += 4) {
        v2f a = *reinterpret_cast<const v2f*>(arow + k0);
        v2f braw = *reinterpret_cast<const v2f*>(brow + k0);
        v2f b; b[0] = braw[0] * bm; b[1] = braw[1] * bm;
        // 8 args: (neg_a, A, neg_b, B, c_mod, C, reuse_a, reuse_b)
        c = __builtin_amdgcn_wmma_f32_16x16x4_f32(false, a, false, b,
                                                  (short)0, c, false, false);
    }

    if (l < NEXP) {
        float* dst = logits + (size_t)(rowBase + 8 * h) * NEXP + l;
#pragma unroll
        for (int r = 0; r < 8; ++r)
            dst[(size_t)r * NEXP] = c[r];   // token rowBase + r + 8h, expert l
    }
}

// ---------------------------------------------------------------------------
// K2: per-token softmax over 8 experts, top-2, renormalize, threshold.
// ---------------------------------------------------------------------------
__global__ void __launch_bounds__(256)
route_kernel(const float* __restrict__ logits, const float* __restrict__ probs,
             float* __restrict__ g, int* __restrict__ eidx, int* __restrict__ r1) {
    const int tok = blockIdx.x * blockDim.x + threadIdx.x;
    if (tok >= NTOK) return;

    float p[NEXP];
#pragma unroll
    for (int e = 0; e < NEXP; ++e) p[e] = logits[(size_t)tok * NEXP + e];

    float m = p[0];
#pragma unroll
    for (int e = 1; e < NEXP; ++e) m = fmaxf(m, p[e]);
    float s = 0.f;
#pragma unroll
    for (int e = 0; e < NEXP; ++e) { p[e] = __expf(p[e] - m); s += p[e]; }
    const float inv = 1.0f / s;
#pragma unroll
    for (int e = 0; e < NEXP; ++e) p[e] *= inv;

    // top-2 (first index wins ties, matching jax top_k)
    float best = -1.f; int bi = 0;
#pragma unroll
    for (int e = 0; e < NEXP; ++e) if (p[e] > best) { best = p[e]; bi = e; }
    float sec = -1.f; int si = 0;
#pragma unroll
    for (int e = 0; e < NEXP; ++e) if (e != bi && p[e] > sec) { sec = p[e]; si = e; }

    float denom = best + sec;
    denom = (denom < 1e-9f) ? 1e-9f : denom;
    const float g0 = best / denom;
    const float g1 = sec  / denom;

    // rank-0 always routes; rank-1: probs[1,b,t] < g1/0.2
    const float pr = probs[NTOK + tok];
    const int route1 = (pr < g1 * 5.0f) ? 1 : 0;

    g[tok * 2 + 0] = g0;  g[tok * 2 + 1] = g1;
    eidx[tok * 2 + 0] = bi; eidx[tok * 2 + 1] = si;
    r1[tok] = route1;
}

// ---------------------------------------------------------------------------
// K3: per-batch exclusive scan over T of per-expert one-hot masks, capacity.
// One block per batch; 256 threads x 8 tokens; Hillis-Steele on 8-vectors.
// ---------------------------------------------------------------------------
#define SCAN_TPB 256
__global__ void __launch_bounds__(SCAN_TPB)
scan_kernel(const int* __restrict__ eidx, const int* __restrict__ r1arr,
            int* __restrict__ pos, int* __restrict__ kept) {
    const int b   = blockIdx.x;
    const int tid = threadIdx.x;
    const int tokBase = b * 2048 + tid * 8;

    __shared__ int buf[2][SCAN_TPB][NEXP];
    __shared__ int keptBase[NEXP];

    // ---------------- rank 0 ----------------
    int local[NEXP]; int myE0[8];
#pragma unroll
    for (int e = 0; e < NEXP; ++e) local[e] = 0;
#pragma unroll
    for (int i = 0; i < 8; ++i) {
        int e = eidx[(tokBase + i) * 2 + 0];
        myE0[i] = e; local[e]++;
    }
#pragma unroll
    for (int e = 0; e < NEXP; ++e) buf[0][tid][e] = local[e];

    int src = 0;
    for (int off = 1; off < SCAN_TPB; off <<= 1) {
        __syncthreads();
#pragma unroll
        for (int e = 0; e < NEXP; ++e) {
            int v = buf[src][tid][e];
            if (tid >= off) v += buf[src][tid - off][e];
            buf[1 - src][tid][e] = v;
        }
        src ^= 1;
    }
    __syncthreads();

    int run[NEXP];
#pragma unroll
    for (int e = 0; e < NEXP; ++e) run[e] = (tid > 0) ? buf[src][tid - 1][e] : 0;
    if (tid == 0) {
#pragma unroll
        for (int e = 0; e < NEXP; ++e)
            keptBase[e] = min(buf[src][SCAN_TPB - 1][e], CAP);
    }
#pragma unroll
    for (int i = 0; i < 8; ++i) {
        int tok = tokBase + i;
        int p0 = run[myE0[i]]++;
        pos[tok * 2 + 0]  = p0;
        kept[tok * 2 + 0] = (p0 < CAP) ? 1 : 0;
    }
    __syncthreads();   // done reading buf (rank 0) before reuse

    // ---------------- rank 1 ----------------
    int myE1[8], myR1[8];
#pragma unroll
    for (int e = 0; e < NEXP; ++e) local[e] = 0;
#pragma unroll
    for (int i = 0; i < 8; ++i) {
        int tok = tokBase + i;
        int e = eidx[tok * 2 + 1];
        int r = r1arr[tok];
        myE1[i] = e; myR1[i] = r;
        local[e] += r;
    }
#pragma unroll
    for (int e = 0; e < NEXP; ++e) buf[0][tid][e] = local[e];

    src = 0;
    for (int off = 1; off < SCAN_TPB; off <<= 1) {
        __syncthreads();
#pragma unroll
        for (int e = 0; e < NEXP; ++e) {
            int v = buf[src][tid][e];
            if (tid >= off) v += buf[src][tid - off][e];
            buf[1 - src][tid][e] = v;
        }
        src ^= 1;
    }
    __syncthreads();
#pragma unroll
    for (int e = 0; e < NEXP; ++e)
        run[e] = ((tid > 0) ? buf[src][tid - 1][e] : 0) + keptBase[e];
#pragma unroll
    for (int i = 0; i < 8; ++i) {
        int tok = tokBase + i;
        if (myR1[i]) {
            int p1 = run[myE1[i]]++;
            pos[tok * 2 + 1]  = p1;
            kept[tok * 2 + 1] = (p1 < CAP) ? 1 : 0;
        } else {
            pos[tok * 2 + 1]  = 0;
            kept[tok * 2 + 1] = 0;
        }
    }
}

// ---------------------------------------------------------------------------
// K4: stream both dense outputs once. One wave per (token, expert) row of 512.
// dispatch = (combine > 0) ? 1 : 0 (straight-through forward value).
// ---------------------------------------------------------------------------
__global__ void __launch_bounds__(256)
write_out_kernel(const float* __restrict__ g, const int* __restrict__ eidx,
                 const int* __restrict__ pos, const int* __restrict__ kept,
                 float* __restrict__ out) {
    const int wave = (blockIdx.x * blockDim.x + threadIdx.x) >> 5;  // 0..65535
    const int lane = threadIdx.x & 31;
    if (wave >= NTOK * NEXP) return;

    const int tok = wave >> 3;
    const int e   = wave & 7;

    const int e0 = eidx[tok * 2 + 0];
    const int e1 = eidx[tok * 2 + 1];
    float v = 0.f; int p = -1;
    if (kept[tok * 2 + 0] && e == e0) { v = g[tok * 2 + 0]; p = pos[tok * 2 + 0]; }
    else if (kept[tok * 2 + 1] && e == e1) { v = g[tok * 2 + 1]; p = pos[tok * 2 + 1]; }

    const size_t base = (size_t)wave * CAP;
    float* disp = out + base;
    float* comb = out + (size_t)OUT_HALF + base;

#pragma unroll
    for (int it = 0; it < 4; ++it) {
        const int c = it * 128 + lane * 4;
        v4f dv, cv;
        dv.x = (c + 0 == p) ? 1.f : 0.f;  cv.x = (c + 0 == p) ? v : 0.f;
        dv.y = (c + 1 == p) ? 1.f : 0.f;  cv.y = (c + 1 == p) ? v : 0.f;
        dv.z = (c + 2 == p) ? 1.f : 0.f;  cv.z = (c + 2 == p) ? v : 0.f;
        dv.w = (c + 3 == p) ? 1.f : 0.f;  cv.w = (c + 3 == p) ? v : 0.f;
        __builtin_nontemporal_store(dv, reinterpret_cast<v4f*>(disp + c));
        __builtin_nontemporal_store(cv, reinterpret_cast<v4f*>(comb + c));
    }
}

// ---------------------------------------------------------------------------
extern "C" void kernel_launch(void* const* d_in, const int* in_sizes, int n_in,
                              void* d_out, int out_size, void* d_ws, size_t ws_size,
                              hipStream_t stream) {
    const float* x     = (const float*)d_in[0];  // [4,2048,1024]
    const float* wg    = (const float*)d_in[1];  // [8,1024]
    const float* probs = (const float*)d_in[2];  // [2,4,2048]
    float* out = (float*)d_out;                  // dispatch ++ combine

    // workspace layout (all 4-byte elements)
    float* logits = (float*)d_ws;                 // 65536 floats
    float* g      = logits + (size_t)NTOK * NEXP; // 16384 floats
    int*   eidx   = (int*)(g + 2 * NTOK);         // 16384 ints
    int*   r1     = eidx + 2 * NTOK;              // 8192 ints
    int*   pos    = r1 + NTOK;                    // 16384 ints
    int*   kept   = pos + 2 * NTOK;               // 16384 ints

    // K1: 512 waves (one per 16-token tile) -> 128 blocks x 128 threads
    gate_gemm_kernel<<<128, 128, 0, stream>>>(x, wg, logits);
    // K2: one thread per token
    route_kernel<<<NTOK / 256, 256, 0, stream>>>(logits, probs, g, eidx, r1);
    // K3: one block per batch
    scan_kernel<<<4, SCAN_TPB, 0, stream>>>(eidx, r1, pos, kept);
    // K4: 65536 waves -> 8192 blocks x 256 threads (8 waves/block)
    write_out_kernel<<<8192, 256, 0, stream>>>(g, eidx, pos, kept, out);
}